// LGCN_matrix_12575664242932
// MI455X (gfx1250) — compile-verified
//
#include <hip/hip_runtime.h>

#define NNODES   150000      // USER_NUM + ITEM_NUM
#define HIDDEN   64
#define NLAYERS  3
#define BATCH    4096

typedef __attribute__((ext_vector_type(2))) float v2f;
typedef __attribute__((ext_vector_type(8))) float v8f;

// Hardware f32 atomic add (global_atomic_add_f32) — avoids CAS loop on the
// L2-resident scatter path of the SpMM.
__device__ __forceinline__ void atomic_add_f32(float* p, float v) {
    unsafeAtomicAdd(p, v);
}

// ---------------------------------------------------------------- zero buffer
__global__ void zero_f4(float4* __restrict__ p, long n4) {
    long i = (long)blockIdx.x * blockDim.x + threadIdx.x;
    long stride = (long)gridDim.x * blockDim.x;
    float4 z = make_float4(0.f, 0.f, 0.f, 0.f);
    for (; i < n4; i += stride) p[i] = z;
}

// ------------------------------------------------- layer-0 gathers + out init
__global__ void init_acc(const float* __restrict__ emb,
                         const int* __restrict__ user, const int* __restrict__ pos,
                         const int* __restrict__ neg,
                         float* __restrict__ ua, float* __restrict__ pa,
                         float* __restrict__ na, float* __restrict__ out) {
    int i = blockIdx.x * blockDim.x + threadIdx.x;
    if (i == 0) out[0] = 0.f;               // d_out is poisoned; stream-ordered
    if (i >= BATCH * HIDDEN) return;
    int b = i >> 6, d = i & 63;
    ua[i] = emb[(long)user[b] * HIDDEN + d];
    pa[i] = emb[(long)pos[b]  * HIDDEN + d];
    na[i] = emb[(long)neg[b]  * HIDDEN + d];
}

// -------------------------------------------------------------- COO SpMM
// One wave32 handles 2 edges: lanes [0,16) -> edge 2w, lanes [16,32) -> edge
// 2w+1. Each lane moves one float4 (16 lanes * 16B = full 256B row): the
// gather of x[col] is a coalesced 256B burst, the scatter is 4 hardware f32
// atomics into the L2-resident output row.
__global__ void spmm_coo(const int* __restrict__ rows, const int* __restrict__ cols,
                         const float* __restrict__ vals,
                         const float* __restrict__ x, float* __restrict__ y, int nnz) {
    long t = (long)blockIdx.x * blockDim.x + threadIdx.x;
    long wave = t >> 5;
    int lane = threadIdx.x & 31;
    long e = wave * 2 + (lane >> 4);
    if (e >= nnz) return;
    int sub = lane & 15;
    int r = rows[e];
    int c = cols[e];
    float v = vals[e];
    float4 d = ((const float4*)(x + (long)c * HIDDEN))[sub];
    float* yp = y + (long)r * HIDDEN + (sub << 2);
    atomic_add_f32(yp + 0, v * d.x);
    atomic_add_f32(yp + 1, v * d.y);
    atomic_add_f32(yp + 2, v * d.z);
    atomic_add_f32(yp + 3, v * d.w);
}

// ----------------------------------------------- accumulate per-layer gathers
__global__ void accum_acc(const float* __restrict__ e,
                          const int* __restrict__ user, const int* __restrict__ pos,
                          const int* __restrict__ neg,
                          float* __restrict__ ua, float* __restrict__ pa,
                          float* __restrict__ na) {
    int i = blockIdx.x * blockDim.x + threadIdx.x;
    if (i >= BATCH * HIDDEN) return;
    int b = i >> 6, d = i & 63;
    ua[i] += e[(long)user[b] * HIDDEN + d];
    pa[i] += e[(long)pos[b]  * HIDDEN + d];
    na[i] += e[(long)neg[b]  * HIDDEN + d];
}

// ------------------------------------------------------- BPR loss via WMMA
// Each wave computes a 16-row tile: D = U(16x64) x P^T(64x16) with a chain of
// 16 V_WMMA_F32_16X16X4_F32 (K=4 each), same for N; the dot products are the
// diagonal of D. f32 WMMA keeps full precision. Launch has no tail, so EXEC
// is all-1s at every WMMA (ISA requirement).
//
// A 16x4 f32 layout: lane L holds M = L&15, K = (L>>4)*2 + {0,1} (2 VGPRs).
// B 4x16 mirrors it (B[k][n] = P[n][k] -> same per-lane addressing).
// C/D 16x16 f32: VGPR r = row r (lanes 0-15) / row r+8 (lanes 16-31), N=lane&15.
// Diagonal: lanes 0..7 -> cp[lane] (rows 0..7); lanes 24..31 -> cp[lane-24]
// (rows 8..15).
__global__ void loss_wmma(const float* __restrict__ ua, const float* __restrict__ pa,
                          const float* __restrict__ na, float* __restrict__ out) {
    int wave = (blockIdx.x * blockDim.x + threadIdx.x) >> 5;  // 256 waves total
    int lane = threadIdx.x & 31;
    int r0 = wave * 16;                 // tile of 16 batch rows
    int m  = lane & 15;
    int kb = (lane >> 4) * 2;
    const float* urow = ua + (long)(r0 + m) * HIDDEN;
    const float* prow = pa + (long)(r0 + m) * HIDDEN;
    const float* nrow = na + (long)(r0 + m) * HIDDEN;

    v8f cp = {};
    v8f cn = {};
#pragma unroll
    for (int kk = 0; kk < 16; ++kk) {
        int k = kk * 4 + kb;
        v2f a;  a.x  = urow[k]; a.y  = urow[k + 1];
        v2f bp; bp.x = prow[k]; bp.y = prow[k + 1];
        v2f bn; bn.x = nrow[k]; bn.y = nrow[k + 1];
        cp = __builtin_amdgcn_wmma_f32_16x16x4_f32(false, a, false, bp,
                                                   (short)0, cp, false, false);
        cn = __builtin_amdgcn_wmma_f32_16x16x4_f32(false, a, false, bn,
                                                   (short)0, cn, false, false);
    }

    // diagonal extraction + loss
    bool valid = (lane < 8) || (lane >= 24);
    if (valid) {
        int r = (lane < 8) ? lane : (lane - 24);
        float dp = cp[r];
        float dn = cn[r];
        // u,p,n are acc/(LAYERS+1)=acc/4  ->  logits scale by 1/16
        float x = (dp - dn) * (1.0f / 16.0f);
        // loss = -log_sigmoid(x) = softplus(-x), numerically stable
        float y = -x;
        float loss = fmaxf(y, 0.0f) + log1pf(__expf(-fabsf(y)));
        atomic_add_f32(out, loss);
    }
}

// ---------------------------------------------------------------------------
extern "C" void kernel_launch(void* const* d_in, const int* in_sizes, int n_in,
                              void* d_out, int out_size, void* d_ws, size_t ws_size,
                              hipStream_t stream) {
    const float* emb  = (const float*)d_in[0];
    const float* vals = (const float*)d_in[1];
    const int*   rows = (const int*)d_in[2];
    const int*   cols = (const int*)d_in[3];
    const int*   user = (const int*)d_in[4];
    const int*   pos  = (const int*)d_in[5];
    const int*   neg  = (const int*)d_in[6];
    int nnz = in_sizes[1];

    // workspace layout: 2 ping-pong embedding buffers + 3 batch accumulators
    float* ws = (float*)d_ws;
    float* e0 = ws;
    float* e1 = e0 + (size_t)NNODES * HIDDEN;
    float* ua = e1 + (size_t)NNODES * HIDDEN;
    float* pa = ua + (size_t)BATCH * HIDDEN;
    float* na = pa + (size_t)BATCH * HIDDEN;
    float* out = (float*)d_out;

    const int acc_threads = BATCH * HIDDEN;               // 262144
    init_acc<<<(acc_threads + 255) / 256, 256, 0, stream>>>(emb, user, pos, neg,
                                                            ua, pa, na, out);

    long n4 = (long)NNODES * HIDDEN / 4;
    long spmm_waves   = ((long)nnz + 1) / 2;
    long spmm_threads = spmm_waves * 32;
    int  spmm_blocks  = (int)((spmm_threads + 255) / 256);

    const float* ein = emb;
    float* ebuf[NLAYERS] = { e0, e1, e0 };
    for (int l = 0; l < NLAYERS; ++l) {
        float* eout = ebuf[l];
        zero_f4<<<2048, 256, 0, stream>>>((float4*)eout, n4);
        spmm_coo<<<spmm_blocks, 256, 0, stream>>>(rows, cols, vals, ein, eout, nnz);
        accum_acc<<<(acc_threads + 255) / 256, 256, 0, stream>>>(eout, user, pos, neg,
                                                                 ua, pa, na);
        ein = eout;
    }

    // 32 blocks * 8 waves * 16 rows = 4096 = BATCH, exact (no tail, EXEC all-1s)
    loss_wmma<<<32, 256, 0, stream>>>(ua, pa, na, out);
}